// BendingEnergyReg_57767310131773
// MI455X (gfx1250) — compile-verified
//
#include <hip/hip_runtime.h>

// ---------------------------------------------------------------------------
// Bending-energy regularizer on a [2,160,160,160,3] f32 grid.
// Bandwidth problem (98.3 MB in, 1 scalar out). CDNA5 strategy:
//   * persistent blocks, 2 tiles each (4394 = 2*2197), DOUBLE-BUFFERED:
//     issue async global->LDS b128 loads (ASYNCcnt) for tile t+1, then
//     s_wait_asynccnt <= 12 to release tile t while t+1's DMA is in flight,
//   * 19-point second-derivative stencil computed from LDS (48 KB/tile,
//     2 buffers = 96 KB of the 320 KB/WGP),
//   * wave32 shuffle + LDS reduction -> one partial per block (d_ws),
//   * deterministic single-block final reduction -> d_out[0].
// 156 = 13*12 and 13*12+4 = 160: tiles cover the volume exactly, no bounds
// checks, and every async b128 transfer is 16B-aligned. Each wave issues
// exactly 12 async instructions per tile (uniform loop), so the wait
// threshold 12 precisely separates the two in-flight tiles.
// ---------------------------------------------------------------------------

#define GX 160
#define NB 2
#define TILE 12                 // output points per block per dim
#define HALO 16                 // TILE + 4 (stencil reaches +-2)
#define NTB 13                  // 156 / TILE
#define NTILES (NB*NTB*NTB*NTB) // 4394
#define NBLOCKS (NTILES/2)      // 2197 persistent blocks, 2 tiles each
// element (float) strides of [B,X,Y,Z,C]
#define SB (GX*GX*GX*3)         // 12,288,000
#define SX (GX*GX*3)            // 76,800
#define SY (GX*3)               // 480
#define ROWF (HALO*3)           // 48 floats = 192 B contiguous per (x,y) row
#define CHUNKS_PER_ROW (ROWF/4) // 12 x b128 per row
#define TILEF (HALO*HALO*ROWF)  // 12288 floats = 48 KB per buffer

// clang ext-vector int4: exact type the async-LDS builtins expect
typedef int v4i __attribute__((ext_vector_type(4)));
typedef __attribute__((address_space(1))) v4i* gptr_v4i;
typedef __attribute__((address_space(3))) v4i* lptr_v4i;

static __device__ __forceinline__ const float* tile_base(const float* g, int t)
{
    const int bz = t % NTB; t /= NTB;
    const int by = t % NTB; t /= NTB;
    const int bx = t % NTB; t /= NTB;
    return g + (size_t)t * SB + (size_t)(bx * TILE) * SX
             + (size_t)(by * TILE) * SY + (size_t)(bz * TILE) * 3;
}

// Issue the 48 KB halo tile transfer: 16x16 rows x 192 B = 3072 b128 chunks,
// 12 per thread (uniform -> 12 async instructions per wave).
static __device__ __forceinline__ void stage_tile(const float* base, float* buf, int tid)
{
#if __has_builtin(__builtin_amdgcn_global_load_async_to_lds_b128)
    for (int j = tid; j < HALO * HALO * CHUNKS_PER_ROW; j += 256) {
        const int sub = j % CHUNKS_PER_ROW;       // 16B chunk within row
        const int row = j / CHUNKS_PER_ROW;       // tx*16 + ty
        const int ty = row % HALO, tx = row / HALO;
        const float* gp = base + (size_t)tx * SX + (size_t)ty * SY + sub * 4;
        float* lp = &buf[row * ROWF + sub * 4];
        __builtin_amdgcn_global_load_async_to_lds_b128(
            (gptr_v4i)gp, (lptr_v4i)lp, /*imm offset*/0, /*cpol*/0);
    }
#else
    for (int j = tid; j < HALO * HALO * CHUNKS_PER_ROW; j += 256) {
        const int sub = j % CHUNKS_PER_ROW;
        const int row = j / CHUNKS_PER_ROW;
        const int ty = row % HALO, tx = row / HALO;
        const float4 v =
            *(const float4*)(base + (size_t)tx * SX + (size_t)ty * SY + sub * 4);
        *(float4*)&buf[row * ROWF + sub * 4] = v;
    }
#endif
}

static __device__ __forceinline__ void wait_async_pending(bool more_in_flight)
{
#if __has_builtin(__builtin_amdgcn_global_load_async_to_lds_b128)
  #if __has_builtin(__builtin_amdgcn_s_wait_asynccnt)
    if (more_in_flight) __builtin_amdgcn_s_wait_asynccnt(12);
    else                __builtin_amdgcn_s_wait_asynccnt(0);
  #else
    if (more_in_flight) asm volatile("s_wait_asynccnt 12" ::: "memory");
    else                asm volatile("s_wait_asynccnt 0"  ::: "memory");
  #endif
#else
    (void)more_in_flight;  // sync fallback: data already in LDS
#endif
}

// 19-point stencil: 6 second derivatives x 3 components, weighted squares.
// Weights per u-component (mixed partials are numerically symmetric).
static __device__ __forceinline__ float compute_tile(const float* buf, int tid)
{
    constexpr float WXX[3] = {1.f, 0.f, 0.f};
    constexpr float WYY[3] = {2.f, 1.f, 0.f};
    constexpr float WZZ[3] = {2.f, 2.f, 1.f};
    constexpr float WXY[3] = {3.f, 1.f, 0.f};
    constexpr float WXZ[3] = {3.f, 2.f, 1.f};
    constexpr float WYZ[3] = {4.f, 3.f, 1.f};

    #define LP(X, Y, Z) (&buf[(((X) * HALO + (Y)) * HALO + (Z)) * 3])
    float acc = 0.f;
    for (int p = tid; p < TILE * TILE * TILE; p += 256) {
        const int oz = p % TILE;
        const int oy = (p / TILE) % TILE;
        const int ox = p / (TILE * TILE);
        const int cx = ox + 2, cy = oy + 2, cz = oz + 2;

        const float* c0   = LP(cx,   cy,   cz  );
        const float* xp   = LP(cx+2, cy,   cz  );
        const float* xm   = LP(cx-2, cy,   cz  );
        const float* yp   = LP(cx,   cy+2, cz  );
        const float* ym   = LP(cx,   cy-2, cz  );
        const float* zp   = LP(cx,   cy,   cz+2);
        const float* zm   = LP(cx,   cy,   cz-2);
        const float* xpyp = LP(cx+1, cy+1, cz  );
        const float* xpym = LP(cx+1, cy-1, cz  );
        const float* xmyp = LP(cx-1, cy+1, cz  );
        const float* xmym = LP(cx-1, cy-1, cz  );
        const float* xpzp = LP(cx+1, cy,   cz+1);
        const float* xpzm = LP(cx+1, cy,   cz-1);
        const float* xmzp = LP(cx-1, cy,   cz+1);
        const float* xmzm = LP(cx-1, cy,   cz-1);
        const float* ypzp = LP(cx,   cy+1, cz+1);
        const float* ypzm = LP(cx,   cy+1, cz-1);
        const float* ymzp = LP(cx,   cy-1, cz+1);
        const float* ymzm = LP(cx,   cy-1, cz-1);

        float t = 0.f;
        #pragma unroll
        for (int c = 0; c < 3; ++c) {
            const float ctr2 = 2.f * c0[c];
            const float dxx = xp[c] - ctr2 + xm[c];
            const float dyy = yp[c] - ctr2 + ym[c];
            const float dzz = zp[c] - ctr2 + zm[c];
            const float dxy = xpyp[c] - xpym[c] - xmyp[c] + xmym[c];
            const float dxz = xpzp[c] - xpzm[c] - xmzp[c] + xmzm[c];
            const float dyz = ypzp[c] - ypzm[c] - ymzp[c] + ymzm[c];
            t += WXX[c]*dxx*dxx + WYY[c]*dyy*dyy + WZZ[c]*dzz*dzz
               + WXY[c]*dxy*dxy + WXZ[c]*dxz*dxz + WYZ[c]*dyz*dyz;
        }
        acc += t;
    }
    #undef LP
    return acc;
}

__global__ __launch_bounds__(256)
void bending_partial(const float* __restrict__ g, float* __restrict__ partial)
{
    __shared__ float tiles[2][TILEF];   // double buffer, 96 KB
    __shared__ float red[8];

    const int tid = threadIdx.x;
    float acc = 0.f;

    int t = blockIdx.x;                 // tile ids: blockIdx.x, blockIdx.x+NBLOCKS
    stage_tile(tile_base(g, t), tiles[0], tid);
    int nb = 0;
    while (true) {
        const int tn = t + NBLOCKS;
        const bool more = (tn < NTILES);
        if (more) stage_tile(tile_base(g, tn), tiles[nb ^ 1], tid);  // prefetch
        wait_async_pending(more);       // release current tile only
        __syncthreads();                // all waves' tile-t data visible
        acc += compute_tile(tiles[nb], tid);
        if (!more) break;
        __syncthreads();                // everyone done reading tiles[nb]
        nb ^= 1;
        t = tn;
    }

    // ---- wave32 + cross-wave reduction ------------------------------------
    #pragma unroll
    for (int off = 16; off > 0; off >>= 1)
        acc += __shfl_down(acc, off, 32);
    if ((tid & 31) == 0) red[tid >> 5] = acc;
    __syncthreads();
    if (tid == 0) {
        float s = 0.f;
        #pragma unroll
        for (int i = 0; i < 8; ++i) s += red[i];
        partial[blockIdx.x] = s;
    }
}

__global__ __launch_bounds__(256)
void bending_final(const float* __restrict__ partial, int n, float* __restrict__ out)
{
    __shared__ float red[8];
    const int tid = threadIdx.x;
    float acc = 0.f;
    for (int i = tid; i < n; i += 256) acc += partial[i];
    #pragma unroll
    for (int off = 16; off > 0; off >>= 1)
        acc += __shfl_down(acc, off, 32);
    if ((tid & 31) == 0) red[tid >> 5] = acc;
    __syncthreads();
    if (tid == 0) {
        float s = 0.f;
        #pragma unroll
        for (int i = 0; i < 8; ++i) s += red[i];
        // each second derivative carried 0.25 -> squared 0.0625; / dxx.size
        constexpr double N = 2.0 * 156.0 * 156.0 * 156.0 * 3.0;
        out[0] = (float)((double)s * (0.0625 / N));
    }
}

extern "C" void kernel_launch(void* const* d_in, const int* in_sizes, int n_in,
                              void* d_out, int out_size, void* d_ws, size_t ws_size,
                              hipStream_t stream)
{
    (void)in_sizes; (void)n_in; (void)out_size; (void)ws_size;
    const float* grid = (const float*)d_in[0];
    float* partials = (float*)d_ws;     // 2197 floats of scratch
    bending_partial<<<NBLOCKS, 256, 0, stream>>>(grid, partials);
    bending_final<<<1, 256, 0, stream>>>(partials, NBLOCKS, (float*)d_out);
}